// Loss_tgm_38036230373449
// MI455X (gfx1250) — compile-verified
//
#include <hip/hip_runtime.h>

typedef float v2f __attribute__((ext_vector_type(2)));
typedef float v4f __attribute__((ext_vector_type(4)));
typedef float v8f __attribute__((ext_vector_type(8)));

#define B_    4
#define N_    32
#define HW_   268324            // 518*518
#define HW4_  67081             // HW_/4 (float4 groups; 518^2 = 4*259^2)
#define G_    (B_ * HW4_)       // total float4 columns = 268324
#define INV_NORM (1.0f / ((float)(N_ - 1) * (float)B_))   // 1/124

// ---------------------------------------------------------------------------
// Wave32 sum via V_WMMA_F32_16X16X4_F32:
//   A (16x4): a.x -> A[L%16, 0 or 2] = lane partial, a.y -> A[:,1/3] = 0
//   B (4x16): all ones  =>  D[m,n] = rowsum(m) = partial(m) + partial(m+16)
//   D vgpr r: lanes 0-15 hold M=r, lanes 16-31 hold M=8+r, so summing the 8
//   components gives half-sums; one shfl_xor(16) completes the 32-lane sum.
// EXEC must be all ones at the call site.
// ---------------------------------------------------------------------------
__device__ __forceinline__ float wave_sum32(float x) {
    v2f a; a.x = x;    a.y = 0.0f;
    v2f b; b.x = 1.0f; b.y = 1.0f;
    v8f c = {};
    c = __builtin_amdgcn_wmma_f32_16x16x4_f32(
            /*neg_a=*/false, a, /*neg_b=*/false, b,
            /*c_mod=*/(short)0, c, /*reuse_a=*/false, /*reuse_b=*/false);
    float s = ((c[0] + c[1]) + (c[2] + c[3])) + ((c[4] + c[5]) + (c[6] + c[7]));
    s += __shfl_xor(s, 16, 32);
    return s;   // every lane holds the full 32-lane sum
}

__device__ __forceinline__ float term4(v4f pc, v4f pp, v4f yc, v4f yp) {
    float s;
    s  = __builtin_fabsf(__builtin_fabsf(pc.x - pp.x) - __builtin_fabsf(yc.x - yp.x));
    s += __builtin_fabsf(__builtin_fabsf(pc.y - pp.y) - __builtin_fabsf(yc.y - yp.y));
    s += __builtin_fabsf(__builtin_fabsf(pc.z - pp.z) - __builtin_fabsf(yc.z - yp.z));
    s += __builtin_fabsf(__builtin_fabsf(pc.w - pp.w) - __builtin_fabsf(yc.w - yp.w));
    return s;
}

// ---------------------------------------------------------------------------
// Kernel 1: each thread owns one float4 pixel-column of one batch, streams the
// 32 frames once (prev frame kept in registers), accumulates
// | |dP| - |dY| |. Block-reduces (WMMA wave sum + LDS) to one partial/block.
// ---------------------------------------------------------------------------
__global__ void __launch_bounds__(256)
tgm_partial_kernel(const float* __restrict__ pred,
                   const float* __restrict__ y,
                   float* __restrict__ partials) {
    __shared__ float lds[8];
    const int g = blockIdx.x * 256 + threadIdx.x;

    float acc = 0.0f;
    if (g < G_) {                       // no early return: keep EXEC=all-1s for WMMA
        const int b  = g / HW4_;
        const int p4 = g - b * HW4_;
        const size_t base = (size_t)b * (size_t)N_ * (size_t)HW_;
        const v4f* __restrict__ P = (const v4f*)(pred + base) + p4;
        const v4f* __restrict__ Y = (const v4f*)(y    + base) + p4;
        const int step = HW4_;          // one frame, in float4 units (1 MB stride)

        v4f pp = __builtin_nontemporal_load(P);
        v4f yp = __builtin_nontemporal_load(Y);
        #pragma unroll 4
        for (int n = 1; n < N_; ++n) {
            P += step; Y += step;
            v4f pc = __builtin_nontemporal_load(P);
            v4f yc = __builtin_nontemporal_load(Y);
            if (n + 1 < N_) {           // pull next frame's lines toward L2
                __builtin_prefetch((const void*)(P + step), 0, 1);
                __builtin_prefetch((const void*)(Y + step), 0, 1);
            }
            acc += term4(pc, pp, yc, yp);
            pp = pc; yp = yc;
        }
    }

    const float w = wave_sum32(acc);            // v_wmma_f32_16x16x4_f32
    const int lane = threadIdx.x & 31;
    const int wid  = threadIdx.x >> 5;
    if (lane == 0) lds[wid] = w;
    __syncthreads();
    if (threadIdx.x == 0) {
        float s = 0.0f;
        #pragma unroll
        for (int i = 0; i < 8; ++i) s += lds[i];
        partials[blockIdx.x] = s;
    }
}

// ---------------------------------------------------------------------------
// Kernel 2: single block deterministically folds the per-block partials and
// applies the 1/((N-1)*B) normalization.
// ---------------------------------------------------------------------------
__global__ void __launch_bounds__(256)
tgm_final_kernel(const float* __restrict__ partials, int n,
                 float* __restrict__ out) {
    __shared__ float lds[8];
    float acc = 0.0f;
    for (int i = threadIdx.x; i < n; i += 256) acc += partials[i];

    const float w = wave_sum32(acc);            // EXEC reconverged: all-1s
    const int lane = threadIdx.x & 31;
    const int wid  = threadIdx.x >> 5;
    if (lane == 0) lds[wid] = w;
    __syncthreads();
    if (threadIdx.x == 0) {
        float s = 0.0f;
        #pragma unroll
        for (int i = 0; i < 8; ++i) s += lds[i];
        out[0] = s * INV_NORM;
    }
}

extern "C" void kernel_launch(void* const* d_in, const int* in_sizes, int n_in,
                              void* d_out, int out_size, void* d_ws, size_t ws_size,
                              hipStream_t stream) {
    const float* pred = (const float*)d_in[0];
    const float* y    = (const float*)d_in[1];
    float* out        = (float*)d_out;
    float* partials   = (float*)d_ws;          // needs blocks*4 B ≈ 4.2 KB

    const int threads = 256;
    const int blocks  = (G_ + threads - 1) / threads;   // 1049

    tgm_partial_kernel<<<blocks, threads, 0, stream>>>(pred, y, partials);
    tgm_final_kernel<<<1, threads, 0, stream>>>(partials, blocks, out);
}